// SessionGraph_90984587199168
// MI455X (gfx1250) — compile-verified
//
#include <hip/hip_runtime.h>
#include <hip/hip_bf16.h>

typedef __attribute__((ext_vector_type(2))) float v2f;
typedef __attribute__((ext_vector_type(8))) float v8f;

#define B_SZ 4096
#define NN   16
#define DD   32
#define LL   20
#define VV   50000
#define VOUT 49999   // V - 1

// ---------------------------------------------------------------------------
// Kernel 1: fused GNN propagation step (STEPS=1) + attention readout -> s_g
// One block per batch element; 512 threads = one per (node n, dim d).
// ---------------------------------------------------------------------------
__global__ __launch_bounds__(512)
void gnn_attn_kernel(const int* __restrict__ alias_inputs,
                     const float* __restrict__ Aadj,
                     const int* __restrict__ items,
                     const int* __restrict__ mask,
                     const float* __restrict__ emb,
                     const float* __restrict__ w_ih,
                     const float* __restrict__ w_hh,
                     const float* __restrict__ b_ih,
                     const float* __restrict__ b_hh,
                     const float* __restrict__ b_iah,
                     const float* __restrict__ b_oah,
                     const float* __restrict__ w_ein,
                     const float* __restrict__ b_ein,
                     const float* __restrict__ w_eout,
                     const float* __restrict__ b_eout,
                     const float* __restrict__ w1,
                     const float* __restrict__ b1,
                     const float* __restrict__ w2,
                     const float* __restrict__ b2,
                     const float* __restrict__ w3,
                     float* __restrict__ sg_out)
{
    __shared__ float h[NN][DD];
    __shared__ float nin[NN][DD];
    __shared__ float nout[NN][DD];
    __shared__ float iin[NN][DD];
    __shared__ float iout[NN][DD];
    __shared__ float seq[LL][DD];
    __shared__ float alpha[LL];
    __shared__ float q1s[DD];

    const int b   = blockIdx.x;
    const int tid = threadIdx.x;
    const int n   = tid >> 5;     // 0..15
    const int d   = tid & 31;     // 0..31

    // h = emb[items]
    h[n][d] = emb[(size_t)items[b * NN + n] * DD + d];
    __syncthreads();

    // node_in = h @ w_ein + b_ein ; node_out = h @ w_eout + b_eout
    float accin  = b_ein[d];
    float accout = b_eout[d];
#pragma unroll
    for (int k = 0; k < DD; ++k) {
        const float hv = h[n][k];
        accin  += hv * w_ein[k * DD + d];
        accout += hv * w_eout[k * DD + d];
    }
    nin[n][d]  = accin;
    nout[n][d] = accout;
    __syncthreads();

    // adjacency message passing
    const float* Ab = Aadj + (size_t)b * NN * (2 * NN);
    float ain  = b_iah[d];
    float aout = b_oah[d];
#pragma unroll
    for (int m = 0; m < NN; ++m) {
        ain  += Ab[n * (2 * NN) + m]      * nin[m][d];
        aout += Ab[n * (2 * NN) + NN + m] * nout[m][d];
    }
    iin[n][d]  = ain;
    iout[n][d] = aout;
    __syncthreads();

    // GRU gates: gi = inputs @ w_ih.T + b_ih ; gh = h @ w_hh.T + b_hh
    float gi_r = b_ih[d], gi_i = b_ih[d + DD], gi_n = b_ih[d + 2 * DD];
#pragma unroll 8
    for (int k = 0; k < DD; ++k) {
        const float x0 = iin[n][k];
        const float x1 = iout[n][k];
        gi_r += x0 * w_ih[(d         ) * 64 + k] + x1 * w_ih[(d         ) * 64 + DD + k];
        gi_i += x0 * w_ih[(d +     DD) * 64 + k] + x1 * w_ih[(d +     DD) * 64 + DD + k];
        gi_n += x0 * w_ih[(d + 2 * DD) * 64 + k] + x1 * w_ih[(d + 2 * DD) * 64 + DD + k];
    }
    float gh_r = b_hh[d], gh_i = b_hh[d + DD], gh_n = b_hh[d + 2 * DD];
#pragma unroll 8
    for (int k = 0; k < DD; ++k) {
        const float hv = h[n][k];
        gh_r += hv * w_hh[(d         ) * DD + k];
        gh_i += hv * w_hh[(d +     DD) * DD + k];
        gh_n += hv * w_hh[(d + 2 * DD) * DD + k];
    }
    const float rg = 1.0f / (1.0f + __expf(-(gi_r + gh_r)));
    const float ig = 1.0f / (1.0f + __expf(-(gi_i + gh_i)));
    const float ng = tanhf(gi_n + rg * gh_n);
    const float hold = h[n][d];
    const float hnew = ng + ig * (hold - ng);
    __syncthreads();
    h[n][d] = hnew;
    __syncthreads();

    // seq[t] = h[alias[b][t]]
    for (int t = n; t < LL; t += NN)
        seq[t][d] = h[alias_inputs[b * LL + t] & (NN - 1)][d];
    __syncthreads();

    // q1 = ht @ w1 + b1 (ht = seq[last])
    if (tid < DD) {
        int s = 0;
#pragma unroll
        for (int t = 0; t < LL; ++t) s += mask[b * LL + t];
        const int lastIdx = s - 1;
        float acc = b1[d];
#pragma unroll
        for (int k = 0; k < DD; ++k) acc += seq[lastIdx][k] * w1[k * DD + d];
        q1s[d] = acc;
    }
    __syncthreads();

    // alpha[t] = sigmoid(q1 + seq[t]@w2 + b2) . w3   (wave-level reduce, wave32)
    for (int t = n; t < LL; t += NN) {
        float q2v = b2[d];
#pragma unroll
        for (int k = 0; k < DD; ++k) q2v += seq[t][k] * w2[k * DD + d];
        const float sg = 1.0f / (1.0f + __expf(-(q1s[d] + q2v)));
        float val = sg * w3[d];
#pragma unroll
        for (int off = 16; off > 0; off >>= 1)
            val += __shfl_xor(val, off, 32);
        if (d == 0) alpha[t] = val;
    }
    __syncthreads();

    // s_g[d] = sum_t alpha[t] * seq[t][d] * mask[t]
    if (tid < DD) {
        float acc = 0.0f;
#pragma unroll
        for (int t = 0; t < LL; ++t)
            acc += alpha[t] * seq[t][d] * (float)mask[b * LL + t];
        sg_out[(size_t)b * DD + d] = acc;
    }
}

// ---------------------------------------------------------------------------
// Kernel 2: scores = s_g @ emb[1:].T   [4096 x 32] x [32 x 49999]
// One wave per 16x16 output tile, V_WMMA_F32_16X16X4_F32 chained 8x over K=32.
// Store-bandwidth bound (819 MB out) -> non-temporal stores.
// A layout (ISA 7.12.2, f32 16x4): VGPR0 = K 0 / K 2 (lane halves), VGPR1 = K 1 / K 3.
// C/D layout: VGPR g -> M = g (lanes 0-15), M = g+8 (lanes 16-31); N = lane%16.
// ---------------------------------------------------------------------------
__global__ __launch_bounds__(256)
void gemm_scores_kernel(const float* __restrict__ sg,
                        const float* __restrict__ emb,
                        float* __restrict__ out)
{
    const int mt   = blockIdx.x;              // 0..255 (M tile)
    const int wave = threadIdx.x >> 5;        // 0..7
    const int lane = threadIdx.x & 31;
    const int lm   = lane & 15;
    const int hi   = lane >> 4;               // lane half
    const int mbase = mt * 16;

    // Load A tile (16 rows x K=32) once per wave: 8 chunks of 16x4.
    v2f A[8];
    const float* sgrow = sg + (size_t)(mbase + lm) * DD;
#pragma unroll
    for (int c = 0; c < 8; ++c) {
        const int k = c * 4 + 2 * hi;
        A[c].x = sgrow[k];
        A[c].y = sgrow[k + 1];
    }

    const int ntiles = (VOUT + 15) / 16;      // 3125
    const int slots  = gridDim.y * 8;

    for (int nt = blockIdx.y * 8 + wave; nt < ntiles; nt += slots) {
        const int nbase = nt * 16;
        const int col   = nbase + lm;
        int erow = col + 1;                   // emb[1:]
        if (erow > VV - 1) erow = VV - 1;     // clamp OOB lanes (store is masked)
        const float* ep = emb + (size_t)erow * DD;

        v8f c = {0.f, 0.f, 0.f, 0.f, 0.f, 0.f, 0.f, 0.f};
#pragma unroll
        for (int ch = 0; ch < 8; ++ch) {
            const int k = ch * 4 + 2 * hi;
            v2f bv;
            bv.x = ep[k];
            bv.y = ep[k + 1];
            c = __builtin_amdgcn_wmma_f32_16x16x4_f32(
                    /*neg_a=*/false, A[ch],
                    /*neg_b=*/false, bv,
                    /*c_mod=*/(short)0, c,
                    /*reuse_a=*/false, /*reuse_b=*/false);
        }

        if (col < VOUT) {
            float* op = out + (size_t)(mbase + 8 * hi) * VOUT + col;
#pragma unroll
            for (int g = 0; g < 8; ++g)
                __builtin_nontemporal_store(c[g], op + (size_t)g * VOUT);
        }
    }
}

// ---------------------------------------------------------------------------
extern "C" void kernel_launch(void* const* d_in, const int* in_sizes, int n_in,
                              void* d_out, int out_size, void* d_ws, size_t ws_size,
                              hipStream_t stream)
{
    const int*   alias_inputs = (const int*)  d_in[0];
    const float* Aadj         = (const float*)d_in[1];
    const int*   items        = (const int*)  d_in[2];
    const int*   mask         = (const int*)  d_in[3];
    const float* emb          = (const float*)d_in[4];
    const float* w_ih         = (const float*)d_in[5];
    const float* w_hh         = (const float*)d_in[6];
    const float* b_ih         = (const float*)d_in[7];
    const float* b_hh         = (const float*)d_in[8];
    const float* b_iah        = (const float*)d_in[9];
    const float* b_oah        = (const float*)d_in[10];
    const float* w_ein        = (const float*)d_in[11];
    const float* b_ein        = (const float*)d_in[12];
    const float* w_eout       = (const float*)d_in[13];
    const float* b_eout       = (const float*)d_in[14];
    const float* w1           = (const float*)d_in[15];
    const float* b1           = (const float*)d_in[16];
    const float* w2           = (const float*)d_in[17];
    const float* b2           = (const float*)d_in[18];
    const float* w3           = (const float*)d_in[19];

    float* sg = (float*)d_ws;                       // [4096, 32] = 512 KB

    gnn_attn_kernel<<<dim3(B_SZ), dim3(512), 0, stream>>>(
        alias_inputs, Aadj, items, mask, emb,
        w_ih, w_hh, b_ih, b_hh, b_iah, b_oah,
        w_ein, b_ein, w_eout, b_eout,
        w1, b1, w2, b2, w3, sg);

    gemm_scores_kernel<<<dim3(256, 64), dim3(256), 0, stream>>>(
        sg, emb, (float*)d_out);
}